// Rankformer_66589172957770
// MI455X (gfx1250) — compile-verified
//
#include <hip/hip_runtime.h>
#include <hip/hip_bf16.h>

#define NU 100000
#define NI 50000
#define NE 1000000
#define DIM 64
#define ALPHA_C 1.0f
#define CLAMP_C 1e-6f

typedef __attribute__((ext_vector_type(2))) float v2f;
typedef __attribute__((ext_vector_type(8))) float v8f;

// ---------------- workspace layout (float offsets, host side) ----------------
constexpr long long ND = (long long)NU * DIM;
constexpr long long MD = (long long)NI * DIM;
constexpr long long o_cnt   = 0;                    // NU  (degree as float)
constexpr long long o_sxi   = o_cnt + NU;           // NU*D
constexpr long long o_svi   = o_sxi + ND;           // NU*D
constexpr long long o_Aacc  = o_svi + ND;           // NU*D
constexpr long long o_P     = o_Aacc + ND;          // NI*D
constexpr long long o_Q     = o_P + MD;             // NI*D
constexpr long long o_Gc    = o_Q + MD;             // NI*D
constexpr long long o_R1    = o_Gc + MD;            // NI
constexpr long long o_R2    = o_R1 + NI;            // NI
constexpr long long o_sxisum= o_R2 + NI;            // D  (sum of xi)
constexpr long long o_svisum= o_sxisum + DIM;       // D  (sum of vi)
constexpr long long o_sxud  = o_svisum + DIM;       // D  (sum xu/duj)
constexpr long long o_svb   = o_sxud + DIM;         // D  (sum vu*(b_pos+a)/duj)
constexpr long long o_sb    = o_svb + DIM;          // 1  (sum (b_pos+a)/duj), padded
constexpr long long o_S     = o_sb + 64;            // D*D
constexpr long long o_T     = o_S + DIM * DIM;      // D*D
constexpr long long ACC_END = o_T + DIM * DIM;      // everything above is memset to 0
constexpr long long o_bpos  = ACC_END;              // NU
constexpr long long o_bneg  = o_bpos + NU;          // NU
constexpr long long o_dden  = o_bneg + NU;          // NU+NI  (d1+d2 per row)
constexpr long long o_xui   = o_dden + (NU + NI);   // NE
constexpr long long o_xn    = o_xui + NE;           // (NU+NI)*D normalized embeddings
constexpr long long o_idui  = o_xn + (long long)(NU + NI) * DIM; // NU: 1/max(cnt,1)
constexpr long long o_iduj  = o_idui + NU;          // NU: 1/max(NI-cnt,1)

__device__ __forceinline__ float waveRed(float v) {
#pragma unroll
  for (int m = 16; m > 0; m >>= 1) v += __shfl_xor(v, m, 32);
  return v;
}

// ---- row-normalize x into xn (one wave32 per 64-float row) ----
__global__ void k_normalize(const float* __restrict__ x, float* __restrict__ xn, int rows) {
  int wid = (blockIdx.x * blockDim.x + threadIdx.x) >> 5;
  int lane = threadIdx.x & 31;
  if (wid >= rows) return;
  int base = wid * DIM + lane * 2;
  float2 v = *(const float2*)(x + base);
  float ss = waveRed(v.x * v.x + v.y * v.y);
  float inv = 1.0f / fmaxf(sqrtf(ss), 1e-12f);
  float2 o{v.x * inv, v.y * inv};
  *(float2*)(xn + base) = o;
}

// ---- per-user degree ----
__global__ void k_degree(const int* __restrict__ u, float* __restrict__ cnt) {
  int e = blockIdx.x * blockDim.x + threadIdx.x;
  if (e < NE) atomicAdd(&cnt[u[e]], 1.0f);
}

// ---- precompute 1/dui, 1/duj once per user (keeps divides out of hot loops) ----
__global__ void k_deg_inv(const float* __restrict__ cnt, float* __restrict__ idui,
                          float* __restrict__ iduj) {
  int uu = blockIdx.x * blockDim.x + threadIdx.x;
  if (uu >= NU) return;
  float c = cnt[uu];
  idui[uu] = 1.0f / fmaxf(c, 1.0f);
  iduj[uu] = 1.0f / fmaxf((float)NI - c, 1.0f);
}

// ---- column sums over items: sum_xi, sum_vi ----
__global__ void k_item_sums(const float* __restrict__ x, const float* __restrict__ xn,
                            float* __restrict__ sxisum, float* __restrict__ svisum) {
  int d = threadIdx.x;  // 0..63
  float a = 0.f, b = 0.f;
  for (int it = blockIdx.x; it < NI; it += gridDim.x) {
    int r = (NU + it) * DIM + d;
    a += xn[r]; b += x[r];
  }
  atomicAdd(&sxisum[d], a);
  atomicAdd(&svisum[d], b);
}

// ---- column sums over users: sum xu/duj, sum vu*(b_pos+a)/duj, scalar sum ----
__global__ void k_user_sums(const float* __restrict__ x, const float* __restrict__ xn,
                            const float* __restrict__ iduj, const float* __restrict__ bpos,
                            float* __restrict__ sxud, float* __restrict__ svb,
                            float* __restrict__ sb) {
  int d = threadIdx.x;
  float a = 0.f, b = 0.f, c = 0.f;
  for (int uu = blockIdx.x; uu < NU; uu += gridDim.x) {
    float inv = iduj[uu];
    int r = uu * DIM + d;
    a += xn[r] * inv;
    float w = (bpos[uu] + ALPHA_C) * inv;
    b += x[r] * w;
    if (d == 0) c += w;
  }
  atomicAdd(&sxud[d], a);
  atomicAdd(&svb[d], b);
  if (d == 0) atomicAdd(sb, c);
}

// ---- edge pass scattering by user: sxi, svi, A; stores xui ----
__global__ void k_edge_u(const int* __restrict__ u, const int* __restrict__ it,
                         const float* __restrict__ x, const float* __restrict__ xn,
                         float* __restrict__ sxi, float* __restrict__ svi,
                         float* __restrict__ Aacc, float* __restrict__ xui) {
  int e = (blockIdx.x * blockDim.x + threadIdx.x) >> 5;
  int lane = threadIdx.x & 31;
  if (e >= NE) return;
  int uu = u[e], ii = it[e];
  int ubase = uu * DIM + lane * 2;
  int ibase = (NU + ii) * DIM + lane * 2;
  float2 xu2 = *(const float2*)(xn + ubase);
  float2 xi2 = *(const float2*)(xn + ibase);
  float2 vi2 = *(const float2*)(x + ibase);
  float dotv = waveRed(xu2.x * xi2.x + xu2.y * xi2.y);
  if (lane == 0) xui[e] = dotv;
  atomicAdd(&sxi[ubase], xi2.x);         atomicAdd(&sxi[ubase + 1], xi2.y);
  atomicAdd(&svi[ubase], vi2.x);         atomicAdd(&svi[ubase + 1], vi2.y);
  atomicAdd(&Aacc[ubase], dotv * vi2.x); atomicAdd(&Aacc[ubase + 1], dotv * vi2.y);
}

// ---- per-user scalars: b_pos, b_neg, denominator (du1==du2) ----
__global__ void k_user_scalars(const float* __restrict__ xn, const float* __restrict__ sxi,
                               const float* __restrict__ sxisum,
                               const float* __restrict__ idui, const float* __restrict__ iduj,
                               float* __restrict__ bpos, float* __restrict__ bneg,
                               float* __restrict__ dden) {
  int uu = (blockIdx.x * blockDim.x + threadIdx.x) >> 5;
  int lane = threadIdx.x & 31;
  if (uu >= NU) return;
  int base = uu * DIM + lane * 2;
  float2 xu2 = *(const float2*)(xn + base);
  float2 s2  = *(const float2*)(sxi + base);
  float sx0 = sxisum[lane * 2], sx1 = sxisum[lane * 2 + 1];
  float pp = waveRed(xu2.x * s2.x + xu2.y * s2.y);
  float pn = waveRed(xu2.x * (sx0 - s2.x) + xu2.y * (sx1 - s2.y));
  if (lane == 0) {
    float bp = pp * idui[uu], bn = pn * iduj[uu];
    bpos[uu] = bp; bneg[uu] = bn;
    dden[uu] = 2.0f * fmaxf(bp - bn + ALPHA_C, CLAMP_C);  // du1 == du2
  }
}

// ---- edge pass scattering by item: P, Q, fused Gc, R1, R2 ----
__global__ void k_edge_i(const int* __restrict__ u, const int* __restrict__ it,
                         const float* __restrict__ x, const float* __restrict__ xn,
                         const float* __restrict__ idui, const float* __restrict__ iduj,
                         const float* __restrict__ bpos, const float* __restrict__ bneg,
                         const float* __restrict__ xui,
                         float* __restrict__ P, float* __restrict__ Q, float* __restrict__ Gc,
                         float* __restrict__ R1, float* __restrict__ R2) {
  int e = (blockIdx.x * blockDim.x + threadIdx.x) >> 5;
  int lane = threadIdx.x & 31;
  if (e >= NE) return;
  int uu = u[e], ii = it[e];
  float ai = idui[uu], aj = iduj[uu];
  float bp = bpos[uu], bn = bneg[uu], xe = xui[e];
  // Gc coefficient: zi1+zi2 edge terms = vu * ((xui-(bn-a))/dui + ((bp+a)-xui)/duj)
  float cg = (xe - (bn - ALPHA_C)) * ai + ((bp + ALPHA_C) - xe) * aj;
  int ubase = uu * DIM + lane * 2;
  int ibase = ii * DIM + lane * 2;
  float2 xu2 = *(const float2*)(xn + ubase);
  float2 vu2 = *(const float2*)(x + ubase);
  atomicAdd(&P[ibase],  xu2.x * ai); atomicAdd(&P[ibase + 1],  xu2.y * ai);
  atomicAdd(&Q[ibase],  xu2.x * aj); atomicAdd(&Q[ibase + 1],  xu2.y * aj);
  atomicAdd(&Gc[ibase], vu2.x * cg); atomicAdd(&Gc[ibase + 1], vu2.y * cg);
  if (lane == 0) {
    atomicAdd(&R1[ii], (ALPHA_C - bn) * ai);
    atomicAdd(&R2[ii], (bp + ALPHA_C) * aj);
  }
}

// ---- WMMA reduction GEMM: C(64x64) += sum_r scale(r)*A[r][:]^T (x) B[r][:] ----
// One wave per block; blockIdx.x = 16 output tiles (4x4), blockIdx.y = K chunk.
// Templated on SCALED so neither specialization branches in the hot loop;
// 32-bit indices so loads lower to sgpr-base + voffset (scale_offset) form.
template <bool SCALED>
__global__ void k_gemm_nt(const float* __restrict__ Arows, const float* __restrict__ Brows,
                          const float* __restrict__ scaleArr,
                          int R, int rowsPerBlock, float* __restrict__ C) {
  int tile = blockIdx.x;
  int tm = (tile >> 2) * 16, tn = (tile & 3) * 16;
  int r0 = blockIdx.y * rowsPerBlock;
  int r1 = r0 + rowsPerBlock; if (r1 > R) r1 = R;
  int lane = threadIdx.x;
  int koff = (lane >> 4) << 1;   // lanes 16..31 hold K=2,3 (ISA 16x4 A/B layout)
  int aoff = tm + (lane & 15);
  int boff = tn + (lane & 15);
  v8f acc0 = {}, acc1 = {};
  for (int r = r0; r < r1; r += 8) {
#pragma unroll
    for (int q = 0; q < 2; ++q) {
      int ra = r + q * 4 + koff, rb = ra + 1;
      v2f a, b;
      a.x = Arows[ra * DIM + aoff];
      a.y = Arows[rb * DIM + aoff];
      if (SCALED) { a.x *= scaleArr[ra]; a.y *= scaleArr[rb]; }
      b.x = Brows[ra * DIM + boff];
      b.y = Brows[rb * DIM + boff];
      if (q == 0)
        acc0 = __builtin_amdgcn_wmma_f32_16x16x4_f32(false, a, false, b, (short)0,
                                                     acc0, false, false);
      else
        acc1 = __builtin_amdgcn_wmma_f32_16x16x4_f32(false, a, false, b, (short)0,
                                                     acc1, false, false);
    }
  }
  acc0 += acc1;
#pragma unroll
  for (int v = 0; v < 8; ++v) {
    int Mloc = (lane < 16) ? v : v + 8;  // C layout: VGPR v -> M=v / M=v+8
    atomicAdd(&C[(tm + Mloc) * DIM + boff - tm], acc0[v]);
  }
}

// ---- users: xu@S via WMMA (16 users x 64 dims per wave) + zu epilogue ----
__global__ void k_user_out(const float* __restrict__ xn, const float* __restrict__ S,
                           const float* __restrict__ Aacc, const float* __restrict__ svi,
                           const float* __restrict__ svisum, const float* __restrict__ bpos,
                           const float* __restrict__ bneg,
                           const float* __restrict__ idui, const float* __restrict__ iduj,
                           const float* __restrict__ dden, float* __restrict__ out) {
  int u0 = blockIdx.x * 16;
  int lane = threadIdx.x;
  int koff = (lane >> 4) << 1;
  int ml = lane & 15;
  v8f acc[4] = {};
  for (int k = 0; k < DIM; k += 4) {
    v2f a;
    int abase = (u0 + ml) * DIM + k + koff;
    a.x = xn[abase];
    a.y = xn[abase + 1];
#pragma unroll
    for (int t = 0; t < 4; ++t) {
      v2f b;
      int bbase = (k + koff) * DIM + t * 16 + ml;
      b.x = S[bbase];
      b.y = S[bbase + DIM];
      acc[t] = __builtin_amdgcn_wmma_f32_16x16x4_f32(false, a, false, b, (short)0,
                                                     acc[t], false, false);
    }
  }
#pragma unroll
  for (int v = 0; v < 8; ++v) {
    int user = u0 + ((lane < 16) ? v : v + 8);
    float ai = idui[user], aj = iduj[user];
    float bn = bneg[user], bp = bpos[user];
    float invdd = 1.0f / dden[user];
#pragma unroll
    for (int t = 0; t < 4; ++t) {
      int d = t * 16 + ml;
      int idx = user * DIM + d;
      float xs = acc[t][v];              // (xu @ S)[user][d]
      float Au = Aacc[idx], sv = svi[idx];
      float svj = svisum[d] - sv;
      float z = (Au - sv * (bn - ALPHA_C)) * ai +
                ((xs - Au) - svj * (bp + ALPHA_C)) * aj;
      out[idx] = z * invdd;
    }
  }
}

// ---- per-item denominator: di1, di2 ----
__global__ void k_item_scalars(const float* __restrict__ xn, const float* __restrict__ P,
                               const float* __restrict__ Q, const float* __restrict__ R1,
                               const float* __restrict__ R2, const float* __restrict__ sxud,
                               const float* __restrict__ sb, float* __restrict__ dden) {
  int ii = (blockIdx.x * blockDim.x + threadIdx.x) >> 5;
  int lane = threadIdx.x & 31;
  if (ii >= NI) return;
  int xbase = (NU + ii) * DIM + lane * 2;
  int pbase = ii * DIM + lane * 2;
  float2 xi2 = *(const float2*)(xn + xbase);
  float2 p2  = *(const float2*)(P + pbase);
  float2 q2  = *(const float2*)(Q + pbase);
  float s0 = sxud[lane * 2], s1 = sxud[lane * 2 + 1];
  float d1 = waveRed(xi2.x * p2.x + xi2.y * p2.y);
  float d2 = waveRed(xi2.x * (s0 - q2.x) + xi2.y * (s1 - q2.y));
  if (lane == 0) {
    float di1 = d1 + R1[ii];
    float di2 = -d2 + (sb[0] - R2[ii]);
    dden[NU + ii] = fmaxf(di1, CLAMP_C) + fmaxf(di2, CLAMP_C);
  }
}

// ---- items: xi@T via WMMA + zi epilogue ----
__global__ void k_item_out(const float* __restrict__ xn, const float* __restrict__ T,
                           const float* __restrict__ Gc, const float* __restrict__ svb,
                           const float* __restrict__ dden, float* __restrict__ out) {
  int i0 = blockIdx.x * 16;
  int lane = threadIdx.x;
  int koff = (lane >> 4) << 1;
  int ml = lane & 15;
  v8f acc[4] = {};
  for (int k = 0; k < DIM; k += 4) {
    v2f a;
    int abase = (NU + i0 + ml) * DIM + k + koff;
    a.x = xn[abase];
    a.y = xn[abase + 1];
#pragma unroll
    for (int t = 0; t < 4; ++t) {
      v2f b;
      int bbase = (k + koff) * DIM + t * 16 + ml;
      b.x = T[bbase];
      b.y = T[bbase + DIM];
      acc[t] = __builtin_amdgcn_wmma_f32_16x16x4_f32(false, a, false, b, (short)0,
                                                     acc[t], false, false);
    }
  }
#pragma unroll
  for (int v = 0; v < 8; ++v) {
    int item = i0 + ((lane < 16) ? v : v + 8);
    int row = NU + item;
    float invdd = 1.0f / dden[row];
#pragma unroll
    for (int t = 0; t < 4; ++t) {
      int d = t * 16 + ml;
      float z = acc[t][v] + Gc[item * DIM + d] - svb[d];
      out[row * DIM + d] = z * invdd;
    }
  }
}

extern "C" void kernel_launch(void* const* d_in, const int* in_sizes, int n_in,
                              void* d_out, int out_size, void* d_ws, size_t ws_size,
                              hipStream_t stream) {
  (void)in_sizes; (void)n_in; (void)out_size; (void)ws_size;
  const float* x = (const float*)d_in[0];
  const int* u  = (const int*)d_in[1];
  const int* it = (const int*)d_in[2];
  float* out = (float*)d_out;
  float* ws  = (float*)d_ws;

  float* cnt   = ws + o_cnt;   float* sxi  = ws + o_sxi;   float* svi  = ws + o_svi;
  float* Aacc  = ws + o_Aacc;  float* P    = ws + o_P;     float* Q    = ws + o_Q;
  float* Gc    = ws + o_Gc;    float* R1   = ws + o_R1;    float* R2   = ws + o_R2;
  float* sxisum= ws + o_sxisum;float* svisum=ws + o_svisum;float* sxud = ws + o_sxud;
  float* svb   = ws + o_svb;   float* sb   = ws + o_sb;    float* Smat = ws + o_S;
  float* Tmat  = ws + o_T;     float* bpos = ws + o_bpos;  float* bneg = ws + o_bneg;
  float* dden  = ws + o_dden;  float* xui  = ws + o_xui;   float* xn   = ws + o_xn;
  float* idui  = ws + o_idui;  float* iduj = ws + o_iduj;

  hipMemsetAsync(d_ws, 0, (size_t)ACC_END * sizeof(float), stream);

  int rows = NU + NI;
  k_normalize<<<(rows * 32 + 255) / 256, 256, 0, stream>>>(x, xn, rows);
  k_degree<<<(NE + 255) / 256, 256, 0, stream>>>(u, cnt);
  k_deg_inv<<<(NU + 255) / 256, 256, 0, stream>>>(cnt, idui, iduj);
  k_item_sums<<<256, 64, 0, stream>>>(x, xn, sxisum, svisum);
  k_edge_u<<<(NE * 32 + 255) / 256, 256, 0, stream>>>(u, it, x, xn, sxi, svi, Aacc, xui);
  k_user_scalars<<<(NU * 32 + 255) / 256, 256, 0, stream>>>(xn, sxi, sxisum, idui, iduj,
                                                            bpos, bneg, dden);
  k_user_sums<<<256, 64, 0, stream>>>(x, xn, iduj, bpos, sxud, svb, sb);
  k_edge_i<<<(NE * 32 + 255) / 256, 256, 0, stream>>>(u, it, x, xn, idui, iduj,
                                                      bpos, bneg, xui, P, Q, Gc, R1, R2);
  // S = vi^T xi  (items, no scale)
  {
    dim3 g(16, 125);
    k_gemm_nt<false><<<g, 32, 0, stream>>>(x + (long long)NU * DIM,
                                           xn + (long long)NU * DIM,
                                           nullptr, NI, 400, Smat);
  }
  // T = (xu/duj)^T vu  (users, scale by 1/duj via cached reciprocals)
  {
    dim3 g(16, 250);
    k_gemm_nt<true><<<g, 32, 0, stream>>>(xn, x, iduj, NU, 400, Tmat);
  }
  k_item_scalars<<<(NI * 32 + 255) / 256, 256, 0, stream>>>(xn, P, Q, R1, R2, sxud, sb, dden);
  k_user_out<<<NU / 16, 32, 0, stream>>>(xn, Smat, Aacc, svi, svisum, bpos, bneg,
                                         idui, iduj, dden, out);
  k_item_out<<<NI / 16, 32, 0, stream>>>(xn, Tmat, Gc, svb, dden, out);
}